// MS_67465346285645
// MI455X (gfx1250) — compile-verified
//
#include <hip/hip_runtime.h>
#include <hip/hip_bf16.h>
#include <float.h>

// ---------------------------------------------------------------------------
// Problem constants (from reference): x is (32, 12, 577, 577) f32.
// score = x[:, :, 0, 1:]  -> per (b,h) a 576-vector; top-24 per head;
// scatter-add ones into count[b][576]; view as 24x24; 3x3 conv, dilation 2,
// pad 1 -> 22x22 = 484 values + bias; output (argsort_desc+1)[:, :24] and count.
// ---------------------------------------------------------------------------
#define S_DIM   577
#define PN      576      // patch_num
#define NHEAD   12
#define TOPK    24
#define HGRID   24       // 24x24 count grid
#define OW      22       // conv output width
#define ON      484      // 22*22 conv outputs
#define NTILES  31       // ceil(484 / 16) WMMA row tiles
#define NBATCH  32

typedef __attribute__((ext_vector_type(2))) float v2f;
typedef __attribute__((ext_vector_type(8))) float v8f;

#if defined(__gfx1250__) && __has_builtin(__builtin_amdgcn_wmma_f32_16x16x4_f32)
#define USE_WMMA 1
#else
#define USE_WMMA 0
#endif

// Padded 24x24 grid read (zero outside).
__device__ __forceinline__ float cpad_get(const float* cnt, int r, int c) {
  return (r >= 0 && r < HGRID && c >= 0 && c < HGRID) ? cnt[r * HGRID + c] : 0.0f;
}

// im2col element P[row][t]: row in [0,484) -> (i,j); tap t in [0,9) -> (a,b);
// input coord (i-1+2a, j-1+2b) of the 24x24 count grid; 0 for padding rows/taps.
__device__ __forceinline__ float pval(const float* cnt, int row, int t) {
  if (row >= ON || t >= 9) return 0.0f;
  const int i = row / OW, j = row % OW;
  const int a = t / 3, b = t % 3;
  return cpad_get(cnt, i - 1 + 2 * a, j - 1 + 2 * b);
}

__global__ __launch_bounds__(NHEAD * 32) void MS_topk_dilconv_rank_kernel(
    const float* __restrict__ x,       // (32,12,577,577)
    const float* __restrict__ conv_w,  // (1,1,3,3)
    const float* __restrict__ conv_b,  // (1,)
    float* __restrict__ out) {         // 32*24 idx-as-float, then 32*484 count
  __shared__ float s_scores[NHEAD * PN]; // per-head score scratch
  __shared__ float s_cnt[PN];            // histogram (viewed as 24x24)
  __shared__ float s_out[ON];            // conv result
  __shared__ float s_scr[ON];            // destructive top-k scratch

  const int b    = blockIdx.x;
  const int tid  = threadIdx.x;
  const int wave = tid >> 5;   // 0..11 == head
  const int lane = tid & 31;

  // Zero histogram.
  for (int i = tid; i < PN; i += blockDim.x) s_cnt[i] = 0.0f;

  // Stage this head's 576 scores into LDS: x[b][wave][0][1:577] (contiguous).
  const float* src = x + ((size_t)(b * NHEAD + wave)) * S_DIM * S_DIM + 1;
  __builtin_prefetch(src, 0, 1);                 // global_prefetch_b8
  volatile float* ws = &s_scores[(size_t)wave * PN];
  for (int e = lane; e < PN; e += 32) ws[e] = src[e];
  __syncthreads();

  // ---- per-head top-24 via 24 iterative wave-argmax passes ----------------
  // Tie-break: smaller index wins (matches stable jax.lax.top_k).
  for (int it = 0; it < TOPK; ++it) {
    float v = -FLT_MAX;
    int idx = PN;
    for (int e = lane; e < PN; e += 32) {        // e % 32 == lane
      float t = ws[e];
      if (t > v) { v = t; idx = e; }             // strict > keeps lowest index
    }
    #pragma unroll
    for (int off = 16; off > 0; off >>= 1) {
      float ov = __shfl_xor(v, off, 32);
      int   oi = __shfl_xor(idx, off, 32);
      if (ov > v || (ov == v && oi < idx)) { v = ov; idx = oi; }
    }
    if (lane == (idx & 31)) ws[idx] = -FLT_MAX;  // owning lane retires winner
    if (lane == 0) atomicAdd(&s_cnt[idx], 1.0f); // ds_add_f32
  }
  __syncthreads();

  // ---- dilated 3x3 conv as Y(484x1) = P(484x9pad12) * w via WMMA ----------
  const float bias = conv_b[0];
  float w9[12];
  #pragma unroll
  for (int t = 0; t < 12; ++t) w9[t] = (t < 9) ? conv_w[t] : 0.0f;

#if USE_WMMA
  {
    const bool hi = (lane >= 16);
    const int  m  = lane & 15;
    // Tile assignment is per-wave uniform -> EXEC all-ones around WMMA.
    for (int T = wave; T < NTILES; T += NHEAD) {
      const int row = T * 16 + m;
      v8f acc = {};
      #pragma unroll
      for (int kb = 0; kb < 3; ++kb) {           // K chunks 0..3, 4..7, 8..11
        const int k0 = kb * 4 + (hi ? 2 : 0);    // A/B layout: half-wave = +K/2
        const int k1 = k0 + 1;
        v2f a;  a.x = pval(s_cnt, row, k0);  a.y = pval(s_cnt, row, k1);
        v2f bf; bf.x = w9[k0];               bf.y = w9[k1];  // bcast over N
        acc = __builtin_amdgcn_wmma_f32_16x16x4_f32(
            /*neg_a=*/false, a, /*neg_b=*/false, bf,
            /*c_mod=*/(short)0, acc, /*reuse_a=*/false, /*reuse_b=*/false);
      }
      // All N columns equal; C/D layout: lane 0 holds M=0..7, lane 16 M=8..15.
      if (m == 0) {
        const int rbase = T * 16 + (hi ? 8 : 0);
        #pragma unroll
        for (int r = 0; r < 8; ++r) {
          const int p = rbase + r;
          if (p < ON) s_out[p] = acc[r] + bias;
        }
      }
    }
  }
#else
  for (int p = tid; p < ON; p += blockDim.x) {
    float acc = bias;
    #pragma unroll
    for (int t = 0; t < 9; ++t) acc += w9[t] * pval(s_cnt, p, t);
    s_out[p] = acc;
  }
#endif
  __syncthreads();

  // ---- stable descending top-24 of the 484 conv outputs (wave 0) ----------
  if (wave == 0) {
    volatile float* vscr = s_scr;
    for (int p = lane; p < ON; p += 32) vscr[p] = s_out[p];
    for (int it = 0; it < TOPK; ++it) {
      float v = -FLT_MAX;
      int idx = ON;
      for (int p = lane; p < ON; p += 32) {
        float t = vscr[p];
        if (t > v) { v = t; idx = p; }
      }
      #pragma unroll
      for (int off = 16; off > 0; off >>= 1) {
        float ov = __shfl_xor(v, off, 32);
        int   oi = __shfl_xor(idx, off, 32);
        if (ov > v || (ov == v && oi < idx)) { v = ov; idx = oi; }
      }
      if (lane == (idx & 31)) vscr[idx] = -FLT_MAX;
      if (lane == 0) out[(size_t)b * TOPK + it] = (float)(idx + 1);
    }
  }

  // ---- emit count (second output, after 32*24 indices) --------------------
  for (int p = tid; p < ON; p += blockDim.x) {
    out[(size_t)NBATCH * TOPK + (size_t)b * ON + p] = s_out[p];
  }
}

extern "C" void kernel_launch(void* const* d_in, const int* in_sizes, int n_in,
                              void* d_out, int out_size, void* d_ws, size_t ws_size,
                              hipStream_t stream) {
  (void)n_in; (void)out_size; (void)d_ws; (void)ws_size;
  const float* x      = (const float*)d_in[0];
  const float* conv_w = (const float*)d_in[1];
  const float* conv_b = (const float*)d_in[2];
  // d_in[3] = select_num (== 24, compile-time TOPK)
  float* out = (float*)d_out;

  const int B = in_sizes[0] / (NHEAD * S_DIM * S_DIM); // 32
  MS_topk_dilconv_rank_kernel<<<dim3(B), dim3(NHEAD * 32), 0, stream>>>(
      x, conv_w, conv_b, out);
}